// DCTChannelAttention_49125835931658
// MI455X (gfx1250) — compile-verified
//
#include <hip/hip_runtime.h>
#include <hip/hip_bf16.h>
#include <math.h>

// Problem constants (from reference)
#define BB   32
#define CC   256
#define HH   64
#define WW   64
#define HWSZ (HH * WW)          // 4096 coefficients per channel
#define TOPK 4                  // P-1

typedef float v2f __attribute__((ext_vector_type(2)));
typedef float v8f __attribute__((ext_vector_type(8)));

// ---------------------------------------------------------------------------
// Kernel 0: build the 64x64 DCT-II matrix D[k][n] = s(k)*cos(pi*(2n+1)k/128)
// ---------------------------------------------------------------------------
__global__ __launch_bounds__(256) void dct_init_kernel(float* __restrict__ dct) {
    int idx = blockIdx.x * 256 + threadIdx.x;        // 16 blocks -> 4096
    int k = idx >> 6;
    int n = idx & 63;
    float s = (k == 0) ? 0.125f : 0.17677669529663687f;   // sqrt(1/64), sqrt(2/64)
    float v = cosf(3.14159265358979323846f * (2.0f * n + 1.0f) * (float)k / 128.0f);
    dct[idx] = v * s;
}

// sorted insert: keep t.x >= t.y >= t.z >= t.w
__device__ __forceinline__ void top4_insert(float v, float4& t) {
    if (v > t.w) {
        t.w = v;
        if (t.w > t.z) { float tmp = t.z; t.z = t.w; t.w = tmp;
            if (t.z > t.y) { tmp = t.y; t.y = t.z; t.z = tmp;
                if (t.y > t.x) { tmp = t.x; t.x = t.y; t.y = tmp; } } }
    }
}

// ---------------------------------------------------------------------------
// Kernel 1: per (b,c) channel: coeff = DH * X * DW^T via f32 WMMA,
//           energy = |DC| + sum(top-4 |AC|).
// One workgroup (8 waves) per channel. 16 tiles of 16x16 per GEMM stage,
// 2 tiles per wave.
// ---------------------------------------------------------------------------
__global__ __launch_bounds__(256) void dct_energy_kernel(
    const float* __restrict__ x,
    const float* __restrict__ dct,
    float* __restrict__ energy)
{
    __shared__ float sD[HWSZ];        // DCT matrix       (16 KB)
    __shared__ float sX[HWSZ];        // input tile       (16 KB)
    __shared__ float sT[HWSZ];        // T = D * X        (16 KB)
    __shared__ float sTop[256 * 4];   // per-thread top-4 ( 4 KB)
    __shared__ float sDC;

    const int tid = threadIdx.x;
    const size_t base = (size_t)blockIdx.x * HWSZ;

    // cooperative 128-bit loads: 1024 float4 each for X and D
    {
        const float4* __restrict__ gx = (const float4*)(x + base);
        const float4* __restrict__ gd = (const float4*)dct;
        float4* lx = (float4*)sX;
        float4* ld = (float4*)sD;
        #pragma unroll
        for (int i = 0; i < 4; ++i) {
            lx[tid + 256 * i] = gx[tid + 256 * i];
            ld[tid + 256 * i] = gd[tid + 256 * i];
        }
    }
    __syncthreads();

    const int wave = tid >> 5;
    const int lane = tid & 31;
    const int l    = lane & 15;   // M (A) / N (B,C) within tile
    const int half = lane >> 4;   // K sub-block selector

    // ---------------- stage 1: T = D * X ----------------
    v8f acc[2];
    #pragma unroll
    for (int t = 0; t < 2; ++t) {
        const int tt = wave + 8 * t;          // tile id 0..15
        const int tm = (tt >> 2) << 4;
        const int tn = (tt & 3)  << 4;
        v8f c = {0.f, 0.f, 0.f, 0.f, 0.f, 0.f, 0.f, 0.f};
        #pragma unroll 4
        for (int kk = 0; kk < 16; ++kk) {
            const int k0 = kk * 4 + half * 2;
            v2f a, b;
            a[0] = sD[(tm + l) * 64 + k0];
            a[1] = sD[(tm + l) * 64 + k0 + 1];
            b[0] = sX[k0 * 64 + tn + l];
            b[1] = sX[(k0 + 1) * 64 + tn + l];
            c = __builtin_amdgcn_wmma_f32_16x16x4_f32(
                    false, a, false, b, (short)0, c, false, false);
        }
        acc[t] = c;
    }
    #pragma unroll
    for (int t = 0; t < 2; ++t) {
        const int tt = wave + 8 * t;
        const int tm = (tt >> 2) << 4;
        const int tn = (tt & 3)  << 4;
        #pragma unroll
        for (int r = 0; r < 8; ++r)
            sT[(tm + r + 8 * half) * 64 + tn + l] = acc[t][r];
    }
    __syncthreads();

    // ---------------- stage 2: coeff = T * D^T, consume on the fly --------
    float4 top = {-1e30f, -1e30f, -1e30f, -1e30f};
    #pragma unroll
    for (int t = 0; t < 2; ++t) {
        const int tt = wave + 8 * t;
        const int tm = (tt >> 2) << 4;
        const int tn = (tt & 3)  << 4;
        v8f c = {0.f, 0.f, 0.f, 0.f, 0.f, 0.f, 0.f, 0.f};
        #pragma unroll 4
        for (int kk = 0; kk < 16; ++kk) {
            const int k0 = kk * 4 + half * 2;
            v2f a, b;
            a[0] = sT[(tm + l) * 64 + k0];
            a[1] = sT[(tm + l) * 64 + k0 + 1];
            b[0] = sD[(tn + l) * 64 + k0];       // B[k][n] = D[n][k]  (D^T)
            b[1] = sD[(tn + l) * 64 + k0 + 1];
            c = __builtin_amdgcn_wmma_f32_16x16x4_f32(
                    false, a, false, b, (short)0, c, false, false);
        }
        #pragma unroll
        for (int r = 0; r < 8; ++r) {
            const int row = tm + r + 8 * half;
            const int col = tn + l;
            const float v = fabsf(c[r]);
            if ((row | col) == 0) sDC = v;       // DC term, exclude from top-k
            else                  top4_insert(v, top);
        }
    }
    sTop[tid * 4 + 0] = top.x;
    sTop[tid * 4 + 1] = top.y;
    sTop[tid * 4 + 2] = top.z;
    sTop[tid * 4 + 3] = top.w;
    __syncthreads();

    // 4-ary tree merge of top-4 lists: 256 -> 64 -> 16 -> 4 -> 1
    for (int width = 64; width >= 1; width >>= 2) {
        float4 m = {-1e30f, -1e30f, -1e30f, -1e30f};
        if (tid < width) {
            #pragma unroll
            for (int s = 0; s < 16; ++s) top4_insert(sTop[tid * 16 + s], m);
        }
        __syncthreads();
        if (tid < width) {
            sTop[tid * 4 + 0] = m.x;
            sTop[tid * 4 + 1] = m.y;
            sTop[tid * 4 + 2] = m.z;
            sTop[tid * 4 + 3] = m.w;
        }
        __syncthreads();
    }

    if (tid == 0)
        energy[blockIdx.x] = sDC + sTop[0] + sTop[1] + sTop[2] + sTop[3];
}

// ---------------------------------------------------------------------------
// Kernel 2: per batch b: hid = relu(energy_b @ w1 + b1) ; attn = sigmoid(hid @ w2 + b2)
// ---------------------------------------------------------------------------
__global__ __launch_bounds__(256) void mlp_attn_kernel(
    const float* __restrict__ energy,   // (B, C)
    const float* __restrict__ w1,       // (C, C/4) = (256, 64)
    const float* __restrict__ b1,       // (64,)
    const float* __restrict__ w2,       // (C/4, C) = (64, 256)
    const float* __restrict__ b2,       // (256,)
    float* __restrict__ attn)           // (B, C)
{
    __shared__ float sE[CC];
    __shared__ float sH[CC / 4];
    const int tid = threadIdx.x;
    const int b   = blockIdx.x;

    sE[tid] = energy[b * CC + tid];
    __syncthreads();

    if (tid < 64) {
        float h = b1[tid];
        #pragma unroll 8
        for (int c = 0; c < CC; ++c) h += sE[c] * w1[c * 64 + tid];
        sH[tid] = fmaxf(h, 0.0f);
    }
    __syncthreads();

    float a = b2[tid];
    #pragma unroll 8
    for (int j = 0; j < 64; ++j) a += sH[j] * w2[j * CC + tid];
    attn[b * CC + tid] = 1.0f / (1.0f + expf(-a));
}

// ---------------------------------------------------------------------------
// Kernel 3: out = x * attn[b,c]   (float4-vectorized, memory bound)
// ---------------------------------------------------------------------------
__global__ __launch_bounds__(256) void scale_kernel(
    const float4* __restrict__ x4,
    const float* __restrict__ attn,
    float4* __restrict__ out4)
{
    const size_t idx = (size_t)blockIdx.x * 256 + threadIdx.x;  // float4 index
    const float a = attn[idx >> 10];            // 4096 floats = 1024 float4 / channel
    float4 v = x4[idx];
    v.x *= a; v.y *= a; v.z *= a; v.w *= a;
    out4[idx] = v;
}

// ---------------------------------------------------------------------------
extern "C" void kernel_launch(void* const* d_in, const int* in_sizes, int n_in,
                              void* d_out, int out_size, void* d_ws, size_t ws_size,
                              hipStream_t stream) {
    const float* x  = (const float*)d_in[0];
    const float* w1 = (const float*)d_in[1];
    const float* b1 = (const float*)d_in[2];
    const float* w2 = (const float*)d_in[3];
    const float* b2 = (const float*)d_in[4];
    float* out = (float*)d_out;

    float* ws     = (float*)d_ws;
    float* dct    = ws;                       // 4096 floats
    float* energy = ws + HWSZ;                // B*C = 8192 floats
    float* attn   = ws + HWSZ + BB * CC;      // B*C = 8192 floats

    dct_init_kernel<<<dim3(HWSZ / 256), dim3(256), 0, stream>>>(dct);
    dct_energy_kernel<<<dim3(BB * CC), dim3(256), 0, stream>>>(x, dct, energy);
    mlp_attn_kernel<<<dim3(BB), dim3(256), 0, stream>>>(energy, w1, b1, w2, b2, attn);
    scale_kernel<<<dim3((BB * CC * HWSZ) / 4 / 256), dim3(256), 0, stream>>>(
        (const float4*)x, attn, (float4*)out);
}